// MultiSpatialAttention_62234076119699
// MI455X (gfx1250) — compile-verified
//
#include <hip/hip_runtime.h>
#include <hip/hip_bf16.h>
#include <stdint.h>

// ---------------------------------------------------------------------------
// MultiSpatialAttention fused kernel for gfx1250 (MI455X).
//   B=32, idf=cdf=256, ih=iw=64 (queryL=4096), sourceL=64, H=8, fp32.
// Memory-bound (~300MB mandatory HBM traffic => ~13us @ 23.3TB/s), so the
// 256MB attn tensor is never written to HBM: logits -> masked softmax ->
// both output contractions are fused, with V_WMMA_F32_16X16X4_F32 doing all
// matrix math. LDS staging uses gfx1250 async global->LDS (ASYNCcnt) when
// the builtins are available, with a vectorized VGPR fallback.
// ---------------------------------------------------------------------------

typedef __attribute__((ext_vector_type(2))) float v2f;
typedef __attribute__((ext_vector_type(4))) float v4f;
typedef __attribute__((ext_vector_type(8))) float v8f;
typedef __attribute__((ext_vector_type(4))) int   v4i;

#if __has_builtin(__builtin_amdgcn_global_load_async_to_lds_b128)
#define MSA_ASYNC_LDS 1
#else
#define MSA_ASYNC_LDS 0
#endif

__device__ __forceinline__ void copy16_to_lds(const float* __restrict__ g,
                                              float* __restrict__ l) {
#if MSA_ASYNC_LDS
    // async DMA: global -> LDS, no VGPR round trip, tracked by ASYNCcnt
    __builtin_amdgcn_global_load_async_to_lds_b128(
        (__attribute__((address_space(1))) v4i*)(const void*)g,
        (__attribute__((address_space(3))) v4i*)(void*)l, 0, 0);
#else
    *(v4f*)l = *(const v4f*)g;
#endif
}

__device__ __forceinline__ void wait_async_lds() {
#if MSA_ASYNC_LDS
#if __has_builtin(__builtin_amdgcn_s_wait_asynccnt)
    __builtin_amdgcn_s_wait_asynccnt(0);
#else
    asm volatile("s_wait_asynccnt 0" ::: "memory");
#endif
#endif
}

__device__ __forceinline__ v8f zero8() {
    v8f z;
#pragma unroll
    for (int i = 0; i < 8; ++i) z[i] = 0.0f;
    return z;
}

__device__ __forceinline__ v8f wmma_f32_k4(v2f a, v2f b, v8f c) {
    // D(16x16,f32) = A(16x4,f32) * B(4x16,f32) + C
    return __builtin_amdgcn_wmma_f32_16x16x4_f32(
        false, a, false, b, (short)0, c, false, false);
}

// problem constants
constexpr int BSZ   = 32;
constexpr int DCH   = 256;   // idf == cdf
constexpr int QL    = 4096;  // ih*iw
constexpr int SRCL  = 64;    // sourceL
constexpr int HEADS = 8;

// tiling
constexpr int THREADS = 128; // 4 waves (wave32)
constexpr int WAVES   = 4;
constexpr int QT      = 64;  // queries per workgroup (16 per wave)

// padded LDS strides (64 banks of 4B)
constexpr int SRC_STRIDE = SRCL + 4;  // 68: rows 16B-aligned for async B128;
                                      // d-major reads -> bank stride 4 (no conflict)
constexpr int ATT_STRIDE = SRCL + 1;  // 65: q-major reads -> bank stride 1

// ---------------------------------------------------------------------------
// Kernel 1: sourceT[b,h] = W[h] (256x256) @ context[b] (256x64)
// grid = (DCH/64, HEADS, B), block = 128. Each wave: 16 d-rows x 64 s.
// ---------------------------------------------------------------------------
__global__ void msa_conv1x1_heads(const float* __restrict__ ctx,  // [B,256,64]
                                  const float* __restrict__ W,    // [H,256,256]
                                  float* __restrict__ srcT)       // [B,H,256,64]
{
    const int dblk = blockIdx.x;          // 0..3 (64 d-rows each)
    const int h    = blockIdx.y;
    const int b    = blockIdx.z;
    const int wave = threadIdx.x >> 5;
    const int lane = threadIdx.x & 31;
    const int ln   = lane & 15;           // N / M index within tile
    const int lh   = lane >> 4;           // half select (K+2 / M+8)

    const int dbase = dblk * 64 + wave * 16;
    const float* __restrict__ Wh = W + (size_t)h * DCH * DCH;
    const float* __restrict__ cb = ctx + (size_t)b * DCH * SRCL;

    v8f acc[4];
#pragma unroll
    for (int t = 0; t < 4; ++t) acc[t] = zero8();

    for (int k0 = 0; k0 < DCH; k0 += 4) {
        const int kc = k0 + 2 * lh;       // this lane's K for VGPR v=0 (v=1 -> +1)
        v2f a;                             // A: M=d (lane), K=c
        a.x = Wh[(size_t)(dbase + ln) * DCH + kc];
        a.y = Wh[(size_t)(dbase + ln) * DCH + kc + 1];
#pragma unroll
        for (int t = 0; t < 4; ++t) {     // B: K=c, N=s
            v2f bm;
            bm.x = cb[(size_t)kc * SRCL + t * 16 + ln];
            bm.y = cb[(size_t)(kc + 1) * SRCL + t * 16 + ln];
            acc[t] = wmma_f32_k4(a, bm, acc[t]);
        }
    }

    // C layout: VGPR r, lanes0-15 -> M=r, lanes16-31 -> M=8+r; N=ln
    float* __restrict__ out = srcT + ((size_t)(b * HEADS + h)) * DCH * SRCL;
#pragma unroll
    for (int t = 0; t < 4; ++t)
#pragma unroll
        for (int r = 0; r < 8; ++r)
            out[(size_t)(dbase + r + 8 * lh) * SRCL + t * 16 + ln] = acc[t][r];
}

// ---------------------------------------------------------------------------
// Kernel 2: fused  logits -> masked softmax -> weightedContext & attnOut
// grid = (QL/QT, B), block = 128. Each wave owns 16 queries.
// ---------------------------------------------------------------------------
__global__ void msa_attn_fused(const float* __restrict__ input,  // [B,256,4096]
                               const float* __restrict__ srcTg,  // [B,H,256,64]
                               const unsigned char* __restrict__ mask, // [B,64]
                               float* __restrict__ wcOut,   // [B,256,4096]
                               float* __restrict__ attOut)  // [B,64,4096]
{
    extern __shared__ float smem[];
    float* __restrict__ tgt  = smem;                       // [256][QT]
    float* __restrict__ srcT = tgt + DCH * QT;             // [256][SRC_STRIDE]
    float* __restrict__ attw = srcT + DCH * SRC_STRIDE;    // [WAVES][16][ATT_STRIDE]

    const int qtile = blockIdx.x;
    const int b     = blockIdx.y;
    const int q0    = qtile * QT;
    const int tid   = threadIdx.x;
    const int wave  = tid >> 5;
    const int lane  = tid & 31;
    const int ln    = lane & 15;
    const int lh    = lane >> 4;
    const int qw    = wave * 16;          // this wave's local query base
    float* __restrict__ aw = attw + wave * 16 * ATT_STRIDE;

    // stage target tile [256 d][QT q] once (16B chunks); reused for all 8 heads
    for (int c = tid; c < DCH * (QT / 4); c += THREADS) {
        const int d = c / (QT / 4), q4 = (c % (QT / 4)) * 4;
        copy16_to_lds(&input[((size_t)b * DCH + d) * QL + q0 + q4],
                      &tgt[d * QT + q4]);
    }

    // per-lane mask registers: s = t*16 + ln
    bool msk[4];
#pragma unroll
    for (int t = 0; t < 4; ++t) msk[t] = mask[b * SRCL + t * 16 + ln] != 0;

    v8f wcacc[16];                        // 256d x 16q, summed over heads
#pragma unroll
    for (int dt = 0; dt < 16; ++dt) wcacc[dt] = zero8();
    v8f atacc[4];                         // 16q x 64s attn sum over heads
#pragma unroll
    for (int t = 0; t < 4; ++t) atacc[t] = zero8();

    for (int h = 0; h < HEADS; ++h) {
        __syncthreads();  // prior head's srcT/tgt readers done
        const float* __restrict__ sg =
            srcTg + ((size_t)(b * HEADS + h)) * DCH * SRCL;
        for (int c = tid; c < DCH * (SRCL / 4); c += THREADS) {
            const int d = c >> 4, s4 = (c & 15) * 4;
            copy16_to_lds(&sg[d * SRCL + s4], &srcT[d * SRC_STRIDE + s4]);
        }
        wait_async_lds();
        __syncthreads();

        // ---- matmul 1: logits (16q x 64s) = targetT @ sourceT, K = 256 ----
        v8f lacc[4];
#pragma unroll
        for (int t = 0; t < 4; ++t) lacc[t] = zero8();
        for (int k0 = 0; k0 < DCH; k0 += 4) {
            const int kd = k0 + 2 * lh;
            v2f a;                        // A: M=q (lane), K=d
            a.x = tgt[(kd + 0) * QT + qw + ln];
            a.y = tgt[(kd + 1) * QT + qw + ln];
#pragma unroll
            for (int t = 0; t < 4; ++t) { // B: K=d, N=s
                v2f bm;
                bm.x = srcT[(kd + 0) * SRC_STRIDE + t * 16 + ln];
                bm.y = srcT[(kd + 1) * SRC_STRIDE + t * 16 + ln];
                lacc[t] = wmma_f32_k4(a, bm, lacc[t]);
            }
        }

        // ---- masked softmax over 64 source positions, per query row ----
        // C row M = r + 8*lh spans lanes (ln=N) and the 4 tiles.
#pragma unroll
        for (int r = 0; r < 8; ++r) {
            float row[4];
#pragma unroll
            for (int t = 0; t < 4; ++t)
                row[t] = msk[t] ? -__builtin_inff() : lacc[t][r];
            float m = fmaxf(fmaxf(row[0], row[1]), fmaxf(row[2], row[3]));
#pragma unroll
            for (int off = 1; off < 16; off <<= 1)
                m = fmaxf(m, __shfl_xor(m, off, 32));
            float sum = 0.0f;
#pragma unroll
            for (int t = 0; t < 4; ++t) { row[t] = __expf(row[t] - m); sum += row[t]; }
#pragma unroll
            for (int off = 1; off < 16; off <<= 1)
                sum += __shfl_xor(sum, off, 32);
#pragma unroll
            for (int t = 0; t < 4; ++t) {
                const float p = row[t] / sum;
                atacc[t][r] += p;                          // attnOut head-sum
                aw[(r + 8 * lh) * ATT_STRIDE + t * 16 + ln] = p;  // transpose buf
            }
        }
        // per-wave LDS region, same-wave DS ops are in-order: no barrier needed

        // ---- matmul 2: wc(256d x 16q) += sourceT @ attnT, K = 64 ----
        for (int k0 = 0; k0 < SRCL; k0 += 4) {
            const int ks = k0 + 2 * lh;
            v2f bm;                       // B: K=s, N=q
            bm.x = aw[ln * ATT_STRIDE + ks + 0];
            bm.y = aw[ln * ATT_STRIDE + ks + 1];
#pragma unroll
            for (int dt = 0; dt < 16; ++dt) {  // A: M=d (lane), K=s
                v2f a;
                a.x = srcT[(dt * 16 + ln) * SRC_STRIDE + ks + 0];
                a.y = srcT[(dt * 16 + ln) * SRC_STRIDE + ks + 1];
                wcacc[dt] = wmma_f32_k4(a, bm, wcacc[dt]);
            }
        }
    }

    // ---- store weightedContextOut [B,256,4096] ----
#pragma unroll
    for (int dt = 0; dt < 16; ++dt)
#pragma unroll
        for (int r = 0; r < 8; ++r) {
            const int d = dt * 16 + r + 8 * lh;
            wcOut[((size_t)b * DCH + d) * QL + q0 + qw + ln] = wcacc[dt][r];
        }

    // ---- store attnOut [B,64,4096] (sum over heads, s-major) ----
#pragma unroll
    for (int t = 0; t < 4; ++t)
#pragma unroll
        for (int r = 0; r < 8; ++r) {
            const int s = t * 16 + ln;
            const int q = q0 + qw + r + 8 * lh;
            attOut[((size_t)b * SRCL + s) * QL + q] = atacc[t][r];
        }
}

// ---------------------------------------------------------------------------
extern "C" void kernel_launch(void* const* d_in, const int* in_sizes, int n_in,
                              void* d_out, int out_size, void* d_ws, size_t ws_size,
                              hipStream_t stream) {
    const float*         input   = (const float*)d_in[0];         // [32,256,64,64]
    const float*         context = (const float*)d_in[1];         // [32,256,64]
    const float*         W       = (const float*)d_in[2];         // [8,256,256]
    const unsigned char* mask    = (const unsigned char*)d_in[3]; // [32,64] bool

    float* wcOut  = (float*)d_out;                             // [32,256,4096]
    float* attOut = wcOut + (size_t)BSZ * DCH * QL;            // [32,64,4096]
    float* srcT   = (float*)d_ws;                              // [32,8,256,64] = 16MB

    dim3 g1(DCH / 64, HEADS, BSZ);   // 4 x 8 x 32
    msa_conv1x1_heads<<<g1, THREADS, 0, stream>>>(context, W, srcT);

    const size_t shmem =
        (size_t)(DCH * QT + DCH * SRC_STRIDE + WAVES * 16 * ATT_STRIDE) * sizeof(float);
    dim3 g2(QL / QT, BSZ);           // 64 x 32
    msa_attn_fused<<<g2, THREADS, shmem, stream>>>(input, srcT, mask, wcOut, attOut);
}